// AUGRU_18528488915491
// MI455X (gfx1250) — compile-verified
//
#include <hip/hip_runtime.h>
#include <math.h>

#define B_ 1024
#define T_ 128
#define H_ 32
#define E_ 32
#define D_ 32
#define A_ 128

typedef float v2f __attribute__((ext_vector_type(2)));
typedef float v8f __attribute__((ext_vector_type(8)));

// ---------------------------------------------------------------------------
// Kernel 1: factored attention GEMM with f32 WMMA, 4 batches per block so the
// 512 KB W1p block is read once per block (L2 traffic 512MB -> 128MB).
//   Meff[g][d,a] = W1x[d,a] + sum_e q[g][e] * W1p[d,e,a]   (LDS, 64 KB)
//   zc[g][a]     = b1[a]    + sum_e q[g][e] * W1q[e, a]
//   z[b,t,a]     = sum_d x[b,t,d] * Meff[g][d,a] + zc[g][a]  (WMMA 16x16x4)
// Fused masked batchnorm partial sums via LDS + global f32 atomics.
// ---------------------------------------------------------------------------
__global__ void __launch_bounds__(256, 1) zgemm_kernel(
    const float* __restrict__ x, const float* __restrict__ query,
    const float* __restrict__ W1, const float* __restrict__ b1,
    const int* __restrict__ lengths,
    float* __restrict__ z, float* __restrict__ statSum, float* __restrict__ statSq)
{
  const int blk  = blockIdx.x;          // handles batches 4*blk .. 4*blk+3
  const int tid  = threadIdx.x;
  const int lane = tid & 31;
  const int wave = tid >> 5;            // 8 waves: wave w owns rows [16w,16w+16)
  const int half = lane >> 4;
  const int l16  = lane & 15;
  const int b0   = 4 * blk;

  __shared__ float q_t[E_ * 4];         // transposed: q_t[e*4 + g]
  __shared__ float Meff[4 * D_ * A_];   // 64 KB
  __shared__ float zc[4 * A_];
  __shared__ float ssum[A_];
  __shared__ float ssq[A_];

  if (tid < 128) {
    const int g = tid & 3, e = tid >> 2;
    q_t[tid] = query[(b0 + g) * E_ + e];
    ssum[tid] = 0.0f; ssq[tid] = 0.0f;
  }

  // Prefetch the x tiles this wave will consume in the WMMA phase.
#pragma unroll
  for (int g = 0; g < 4; ++g)
    __builtin_prefetch(x + (((size_t)(b0 + g)) * T_ + 16 * wave + l16) * D_ + 2 * half, 0, 0);

  __syncthreads();

  // zc: rows 1056..1087 of W1 are the q block. 512 entries, 2 per thread.
#pragma unroll
  for (int i = 0; i < 2; ++i) {
    const int idx = tid + 256 * i;      // g*128 + a
    const int g = idx >> 7, a = idx & 127;
    float acc = b1[a];
    const float* wq = W1 + (size_t)(D_ + D_ * E_) * A_ + a;
#pragma unroll 8
    for (int e = 0; e < E_; ++e) acc += q_t[e * 4 + g] * wq[e * A_];
    zc[idx] = acc;
  }

  // Meff: 4096 (d,a) entries, 16 per thread; each W1p element loaded once and
  // reused for 4 batches (q fetched as one ds_load_b128).
#pragma unroll 1
  for (int i = 0; i < 16; ++i) {
    const int idx = tid + 256 * i;      // d*128 + a
    const int d = idx >> 7, a = idx & 127;
    const float base = W1[d * A_ + a];  // W1x block
    float a0 = base, a1 = base, a2 = base, a3 = base;
    const float* wp = W1 + (size_t)(D_ + d * E_) * A_ + a;
#pragma unroll 8
    for (int e = 0; e < E_; ++e) {
      const float w = wp[e * A_];
      const float4 qv = *(const float4*)&q_t[e * 4];
      a0 = fmaf(qv.x, w, a0); a1 = fmaf(qv.y, w, a1);
      a2 = fmaf(qv.z, w, a2); a3 = fmaf(qv.w, w, a3);
    }
    Meff[0 * 4096 + idx] = a0; Meff[1 * 4096 + idx] = a1;
    Meff[2 * 4096 + idx] = a2; Meff[3 * 4096 + idx] = a3;
  }
  __syncthreads();

#pragma unroll 1
  for (int g = 0; g < 4; ++g) {
    const int b = b0 + g;
    const float* xp = x + ((size_t)b * T_ + 16 * wave + l16) * D_ + 2 * half;
    v2f afrag[8];
#pragma unroll
    for (int ki = 0; ki < 8; ++ki) afrag[ki] = *(const v2f*)(xp + 4 * ki);
    const int len = lengths[b];
    const float* Mg = Meff + g * 4096;
    const float* zg = zc + g * A_;

#pragma unroll 1
    for (int n = 0; n < 8; ++n) {       // 8 column tiles of 16
      v8f c = {};
#pragma unroll
      for (int ki = 0; ki < 8; ++ki) {  // K = 32 in steps of 4
        v2f bfrag;
        bfrag.x = Mg[(4 * ki + 2 * half + 0) * A_ + 16 * n + l16];
        bfrag.y = Mg[(4 * ki + 2 * half + 1) * A_ + 16 * n + l16];
        c = __builtin_amdgcn_wmma_f32_16x16x4_f32(false, afrag[ki], false, bfrag,
                                                  (short)0, c, false, false);
      }
      const int colg = 16 * n + l16;
      const float zcv = zg[colg];
      float lsum = 0.0f, lsq = 0.0f;
#pragma unroll
      for (int r = 0; r < 8; ++r) {
        const int row = 16 * wave + r + 8 * half;   // C layout: M=r / M=r+8
        const float val = c[r] + zcv;
        z[((size_t)b * T_ + row) * A_ + colg] = val;
        const float m = (row < len) ? 1.0f : 0.0f;  // mask-multiply, no branch chain
        lsum = fmaf(m, val, lsum);
        lsq  = fmaf(m * val, val, lsq);
      }
      atomicAdd(&ssum[colg], lsum);
      atomicAdd(&ssq[colg], lsq);
    }
  }
  __syncthreads();
  if (tid < A_) {
    atomicAdd(&statSum[tid], ssum[tid]);
    atomicAdd(&statSq[tid],  ssq[tid]);
  }
}

// ---------------------------------------------------------------------------
// Kernel 2: finalize masked batchnorm stats. n = sum(lengths).
// ---------------------------------------------------------------------------
__global__ void __launch_bounds__(128) stats_kernel(
    const float* __restrict__ statSum, const float* __restrict__ statSq,
    const int* __restrict__ lengths,
    float* __restrict__ meanOut, float* __restrict__ invstdOut)
{
  __shared__ float red[128];
  const int tid = threadIdx.x;
  float p = 0.0f;
  for (int i = tid; i < B_; i += 128) p += (float)lengths[i];
  red[tid] = p;
  __syncthreads();
  for (int s = 64; s > 0; s >>= 1) {
    if (tid < s) red[tid] += red[tid + s];
    __syncthreads();
  }
  const float n = red[0];
  const float mean = statSum[tid] / n;
  const float var  = statSq[tid] / n - mean * mean;
  meanOut[tid]   = mean;
  invstdOut[tid] = rsqrtf(fmaxf(var, 0.0f) + 1e-8f);
}

// ---------------------------------------------------------------------------
// Kernel 3: att[b,t] = (z*(ps + (1-ps)*alpha)) . W2 + b2, one wave per row.
// ---------------------------------------------------------------------------
__global__ void __launch_bounds__(256) att_kernel(
    const float* __restrict__ z, const float* __restrict__ mean,
    const float* __restrict__ invstd, const float* __restrict__ alpha,
    const float* __restrict__ W2, const float* __restrict__ b2,
    float* __restrict__ att)
{
  const int row  = blockIdx.x * 8 + (threadIdx.x >> 5);   // b*T + t
  const int lane = threadIdx.x & 31;
  const float4 zv = ((const float4*)(z + (size_t)row * A_))[lane];
  const float zs[4] = {zv.x, zv.y, zv.z, zv.w};
  float acc = 0.0f;
#pragma unroll
  for (int c = 0; c < 4; ++c) {
    const int a = lane * 4 + c;
    const float zn = (zs[c] - mean[a]) * invstd[a];
    const float ps = 1.0f / (1.0f + expf(-zn));
    const float dv = zs[c] * (ps + (1.0f - ps) * alpha[a]);
    acc += dv * W2[a];
  }
#pragma unroll
  for (int m = 16; m > 0; m >>= 1) acc += __shfl_xor(acc, m, 32);
  if (lane == 0) att[row] = acc + b2[0];
}

// ---------------------------------------------------------------------------
// Kernel 3b: hoist everything x-dependent out of the serial recurrence:
//   Cx[row, 0:32] = x[row] @ Wc[0:32, :]
//   gu[row]       = x[row] . Wu[0:32]
//   gr[row]       = x[row] . Wr[0:32]
// One WMMA GEMM with B = [Wc | Wu | Wr | 0-pad] (32 x 48) in LDS.
// ---------------------------------------------------------------------------
__global__ void __launch_bounds__(256) xform_kernel(
    const float* __restrict__ x, const float* __restrict__ Wc,
    const float* __restrict__ Wu, const float* __restrict__ Wr,
    float* __restrict__ Cx, float* __restrict__ gu, float* __restrict__ gr)
{
  const int tid  = threadIdx.x;
  const int lane = tid & 31;
  const int wave = tid >> 5;
  const int half = lane >> 4;
  const int l16  = lane & 15;

  __shared__ float Bm[D_ * 48];         // [k*48 + j]
#pragma unroll
  for (int i = 0; i < 6; ++i) {
    const int idx = tid + 256 * i;
    const int k = idx / 48, j = idx % 48;
    float v = 0.0f;
    if (j < 32)       v = Wc[k * H_ + j];
    else if (j == 32) v = Wu[k];
    else if (j == 33) v = Wr[k];
    Bm[idx] = v;
  }
  __syncthreads();

  const int rowbase = blockIdx.x * 128 + 16 * wave;     // flat row = b*T + t
  const float* xp = x + (size_t)(rowbase + l16) * D_ + 2 * half;
  v2f afrag[8];
#pragma unroll
  for (int ki = 0; ki < 8; ++ki) afrag[ki] = *(const v2f*)(xp + 4 * ki);

#pragma unroll 1
  for (int n = 0; n < 3; ++n) {
    v8f c = {};
#pragma unroll
    for (int ki = 0; ki < 8; ++ki) {
      v2f bfrag;
      bfrag.x = Bm[(4 * ki + 2 * half + 0) * 48 + 16 * n + l16];
      bfrag.y = Bm[(4 * ki + 2 * half + 1) * 48 + 16 * n + l16];
      c = __builtin_amdgcn_wmma_f32_16x16x4_f32(false, afrag[ki], false, bfrag,
                                                (short)0, c, false, false);
    }
#pragma unroll
    for (int r = 0; r < 8; ++r) {
      const int row = rowbase + r + 8 * half;
      const float val = c[r];
      if (n < 2) {
        Cx[(size_t)row * H_ + 16 * n + l16] = val;
      } else if (l16 == 0) {
        gu[row] = val;
      } else if (l16 == 1) {
        gr[row] = val;
      }
    }
  }
}

// ---------------------------------------------------------------------------
// Kernel 4: AUGRU recurrence. One wave32 per batch, lane = hidden index.
// x-dependent terms precomputed; per step only h-dependent work remains.
// ---------------------------------------------------------------------------
__global__ void __launch_bounds__(32) augru_kernel(
    const float* __restrict__ Cx, const float* __restrict__ gu,
    const float* __restrict__ gr, const float* __restrict__ att,
    const int* __restrict__ lengths,
    const float* __restrict__ Wu, const float* __restrict__ bu,
    const float* __restrict__ Wr, const float* __restrict__ br,
    const float* __restrict__ Wc, const float* __restrict__ bc,
    float* __restrict__ out, float* __restrict__ hlast)
{
  const int b    = blockIdx.x;
  const int lane = threadIdx.x;

  const float wuh = Wu[H_ + lane];
  const float wrh = Wr[H_ + lane];
  const float buv = bu[0], brv = br[0];
  const float bcl = bc[lane];

  float wch[32];                        // Wc[H_+k, lane]
#pragma unroll
  for (int k = 0; k < 32; ++k) wch[k] = Wc[(H_ + k) * H_ + lane];

  const float* cxb = Cx  + (size_t)b * T_ * H_;
  const float* gub = gu  + (size_t)b * T_;
  const float* grb = gr  + (size_t)b * T_;
  const float* ab  = att + (size_t)b * T_;
  float*       ob  = out + (size_t)b * T_ * H_;
  const int len = lengths[b];

  float h = 0.0f;
  for (int t = 0; t < T_; ++t) {
    const float cx  = cxb[t * H_ + lane];
    const float guv = gub[t];
    const float grv = grb[t];
    const float a_t = ab[t];
    float su = h * wuh;
    float sr = h * wrh;
#pragma unroll
    for (int m = 16; m > 0; m >>= 1) {
      su += __shfl_xor(su, m, 32);
      sr += __shfl_xor(sr, m, 32);
    }
    const float u = (1.0f / (1.0f + expf(-(guv + su + buv)))) * a_t;
    const float r =  1.0f / (1.0f + expf(-(grv + sr + brv)));
    float acc2 = 0.0f;
#pragma unroll
    for (int k = 0; k < 32; ++k) acc2 = fmaf(__shfl(h, k, 32), wch[k], acc2);
    const float cand = tanhf(cx + bcl + r * acc2);
    const float hn = (1.0f - u) * h + u * cand;
    const bool act = t < len;
    ob[t * H_ + lane] = act ? hn : 0.0f;
    h = act ? hn : h;
  }
  hlast[b * H_ + lane] = h;
}

// ---------------------------------------------------------------------------
extern "C" void kernel_launch(void* const* d_in, const int* in_sizes, int n_in,
                              void* d_out, int out_size, void* d_ws, size_t ws_size,
                              hipStream_t stream) {
  const float* x       = (const float*)d_in[0];
  const float* query   = (const float*)d_in[1];
  const int*   lengths = (const int*)  d_in[2];
  const float* Wu      = (const float*)d_in[3];
  const float* bu      = (const float*)d_in[4];
  const float* Wr      = (const float*)d_in[5];
  const float* br      = (const float*)d_in[6];
  const float* Wc      = (const float*)d_in[7];
  const float* bc      = (const float*)d_in[8];
  const float* W1      = (const float*)d_in[9];
  const float* b1      = (const float*)d_in[10];
  const float* alpha   = (const float*)d_in[11];
  const float* W2      = (const float*)d_in[12];
  const float* b2      = (const float*)d_in[13];

  float* out   = (float*)d_out;                    // (B,T,H) then (B,H)
  float* hlast = out + (size_t)B_ * T_ * H_;

  // Workspace layout (floats)
  const size_t Z_ELEMS = (size_t)B_ * T_ * A_;     // 16M floats = 64 MB
  float* z       = (float*)d_ws;
  float* statSum = z + Z_ELEMS;
  float* statSq  = statSum + A_;
  float* meanv   = statSum + 2 * A_;
  float* invstd  = statSum + 3 * A_;
  float* att     = statSum + 4 * A_;               // B*T floats
  float* Cx      = att + (size_t)B_ * T_;          // B*T*H floats (16 MB)
  float* gu      = Cx + (size_t)B_ * T_ * H_;      // B*T floats
  float* gr      = gu + (size_t)B_ * T_;           // B*T floats

  hipMemsetAsync(statSum, 0, 2 * A_ * sizeof(float), stream);

  zgemm_kernel<<<B_ / 4, 256, 0, stream>>>(x, query, W1, b1, lengths, z, statSum, statSq);
  stats_kernel<<<1, 128, 0, stream>>>(statSum, statSq, lengths, meanv, invstd);
  att_kernel<<<(B_ * T_) / 8, 256, 0, stream>>>(z, meanv, invstd, alpha, W2, b2, att);
  xform_kernel<<<(B_ * T_) / 128, 256, 0, stream>>>(x, Wc, Wu, Wr, Cx, gu, gr);
  augru_kernel<<<B_, 32, 0, stream>>>(Cx, gu, gr, att, lengths,
                                      Wu, bu, Wr, br, Wc, bc, out, hlast);
}